// CustomLoss_29738353557664
// MI455X (gfx1250) — compile-verified
//
#include <hip/hip_runtime.h>
#include <cstdint>

// Problem constants (match reference)
#define B_ 8
#define C_ 512
#define T_ 2048
#define H_ 4

#define WAVES 4                   // waves per block, each owns C_/WAVES classes
#define CPW (C_ / WAVES)          // 128 classes per wave
#define CHUNK 16                  // classes staged per chunk (2 KB per chunk)
#define NCHUNK (CPW / CHUNK)      // 8 chunks per wave
#define NBUF 4                    // LDS ring buffers per wave
#define DEPTH 3                   // async chunks issued ahead (6 KB in flight/wave)

#define COLS (B_ * T_)            // 16384 (b,t) columns
#define BLOCKS_X (COLS / 32)      // 512 blocks per head (32 columns each)
#define NPART (BLOCKS_X * H_)     // 2048 partial sums

// Block = 4 waves x 32 lanes. Lane <-> t-column; wave <-> class slice.
// Online softmax per (wave, column) with triple-ahead async global->LDS
// staging (ASYNCcnt), then a log-sum-exp merge of the 4 class-slice partials.
// Heads 1..3 skip the whole block when no column's parent label matches.
__global__ __launch_bounds__(128)
void hce_loss_kernel(const float* __restrict__ l0, const float* __restrict__ l1,
                     const float* __restrict__ l2, const float* __restrict__ l3,
                     const long long* __restrict__ y, float* __restrict__ ws) {
  // 4 waves x 4 buffers x (16 classes x 32 t) fp32 = 32 KB, wave-private rings
  __shared__ __align__(16) float sbuf[WAVES * NBUF * CHUNK * 32];
  __shared__ float cm[WAVES][32];   // per-wave running max per column
  __shared__ float cs[WAVES][32];   // per-wave rescaled exp-sum per column

  const int lane = threadIdx.x & 31;
  const int wave = threadIdx.x >> 5;
  const int head = blockIdx.y;
  const int col0 = blockIdx.x * 32;       // block's first (b,t) column
  const int b    = col0 >> 11;            // T_ == 2048
  const int t0   = col0 & (T_ - 1);
  const int t    = t0 + lane;

  const long long* yt = y + (((size_t)b * T_ + t) * H_);
  const int parent = (int)yt[0];
  const int label  = (int)yt[head];

  const float* Lg = (head == 0) ? l0 : (head == 1) ? l1 : (head == 2) ? l2 : l3;

  const bool active = (head == 0) || (parent == head - 1);
  // Same 32 columns in every wave -> ballot is block-uniform: safe to branch.
  if (__ballot(active) != 0ull) {
    // This wave's class slice: [wave*CPW, wave*CPW + CPW)
    const float* baseb = Lg + (size_t)b * C_ * T_;          // block base
    const float* basew = baseb + (size_t)wave * CPW * T_;   // wave slice base
    const unsigned long long base_u64 = (unsigned long long)(uintptr_t)basew;

    // Per-lane mapping for one b128 async op: 4 class rows x (8 lanes * 16B)
    const int cl = lane >> 3;               // class row within quad (0..3)
    const int tl = (lane & 7) * 4;          // t offset (4 floats per lane)
    const uint32_t g0   = (uint32_t)(((cl * T_) + t0 + tl) * 4);  // byte off
    const uint32_t lds0 = (uint32_t)(uintptr_t)
        (&sbuf[wave * (NBUF * CHUNK * 32) + cl * 32 + tl]);

    // Stage chunk k (16 classes x 32 t) into ring buffer (k&3): 4 async b128s
    auto issue = [&](int k) {
      const uint32_t g = g0 + (uint32_t)(k * (CHUNK * T_ * 4));
      const uint32_t l = lds0 + (uint32_t)((k & (NBUF - 1)) * (CHUNK * 32 * 4));
#pragma unroll
      for (int q = 0; q < 4; ++q) {
        asm volatile("global_load_async_to_lds_b128 %0, %1, %2"
                     :: "v"(l + (uint32_t)(q * 4 * 32 * 4)),
                        "v"(g + (uint32_t)(q * 4 * T_ * 4)),
                        "s"(base_u64)
                     : "memory");
      }
    };

    float m = -3.402823466e38f, s = 0.0f;
    issue(0); issue(1); issue(2);           // DEPTH == 3 prologue
    for (int k = 0; k < NCHUNK; ++k) {
      if (k + DEPTH < NCHUNK) {
        // reads of the buffer being overwritten (iteration k-1) are retired
        asm volatile("s_wait_dscnt 0x0" ::: "memory");
        issue(k + DEPTH);
        asm volatile("s_wait_asynccnt 0xc" ::: "memory");  // chunk k landed
      } else if (k + 2 < NCHUNK) {
        asm volatile("s_wait_asynccnt 0x8" ::: "memory");
      } else if (k + 1 < NCHUNK) {
        asm volatile("s_wait_asynccnt 0x4" ::: "memory");
      } else {
        asm volatile("s_wait_asynccnt 0x0" ::: "memory");
      }
      const float* buf =
          &sbuf[wave * (NBUF * CHUNK * 32) + (k & (NBUF - 1)) * (CHUNK * 32)];
      float x[CHUNK];
#pragma unroll
      for (int j = 0; j < CHUNK; ++j) x[j] = buf[j * 32 + lane]; // conflict-free
      float mc = x[0];
#pragma unroll
      for (int j = 1; j < CHUNK; ++j) mc = fmaxf(mc, x[j]);
      const float mn = fmaxf(m, mc);
      s *= __expf(m - mn);                  // rescale running sum
#pragma unroll
      for (int j = 0; j < CHUNK; ++j) s += __expf(x[j] - mn);
      m = mn;
    }

    cm[wave][lane] = m;
    cs[wave][lane] = s;
    __syncthreads();

    if (wave == 0) {
      // log-sum-exp merge of the 4 class-slice partials for this column
      float M = cm[0][lane];
#pragma unroll
      for (int w = 1; w < WAVES; ++w) M = fmaxf(M, cm[w][lane]);
      float S = 0.0f;
#pragma unroll
      for (int w = 0; w < WAVES; ++w) S += cs[w][lane] * __expf(cm[w][lane] - M);
      const float lse  = M + __logf(S);
      const float xlab = baseb[(size_t)label * T_ + t];
      const float nll  = lse - xlab;
      const float wgt = (head == 0) ? (1.0f / (float)(B_ * T_))
                                    : (active ? (1.0f / (float)B_) : 0.0f);
      float v = wgt * nll;
#pragma unroll
      for (int off = 16; off > 0; off >>= 1) v += __shfl_xor(v, off, 32);
      if (lane == 0) ws[head * BLOCKS_X + blockIdx.x] = v;
    }
  } else {
    // gated-out block: still must write its partial slot (ws is poisoned)
    if (threadIdx.x == 0) ws[head * BLOCKS_X + blockIdx.x] = 0.0f;
  }
}

__global__ __launch_bounds__(256)
void hce_reduce_kernel(const float* __restrict__ ws, float* __restrict__ out) {
  float v = 0.0f;
  for (int i = threadIdx.x; i < NPART; i += 256) v += ws[i];
#pragma unroll
  for (int off = 16; off > 0; off >>= 1) v += __shfl_xor(v, off, 32);
  __shared__ float sr[8];
  if ((threadIdx.x & 31) == 0) sr[threadIdx.x >> 5] = v;
  __syncthreads();
  if (threadIdx.x == 0) {
    float acc = 0.0f;
#pragma unroll
    for (int i = 0; i < 8; ++i) acc += sr[i];
    out[0] = acc;
  }
}

extern "C" void kernel_launch(void* const* d_in, const int* in_sizes, int n_in,
                              void* d_out, int out_size, void* d_ws, size_t ws_size,
                              hipStream_t stream) {
  (void)in_sizes; (void)n_in; (void)out_size; (void)ws_size;
  const float* l0 = (const float*)d_in[0];
  const float* l1 = (const float*)d_in[1];
  const float* l2 = (const float*)d_in[2];
  const float* l3 = (const float*)d_in[3];
  const long long* y = (const long long*)d_in[4];  // int64 labels
  float* ws  = (float*)d_ws;                       // 2048 fp32 partials (8 KB)
  float* out = (float*)d_out;                      // scalar fp32

  dim3 grid(BLOCKS_X, H_);     // 512 column-tiles x 4 heads, 4 waves each
  hce_loss_kernel<<<grid, dim3(128), 0, stream>>>(l0, l1, l2, l3, y, ws);
  hce_reduce_kernel<<<1, 256, 0, stream>>>(ws, out);
}